// CustomEmbedding_19078244728842
// MI455X (gfx1250) — compile-verified
//
#include <hip/hip_runtime.h>

// Embedding gather with override table.
//   out[tok, :] = (ids[tok] >= over_start) ? extra[ids[tok]-over_start, :]
//                                          : weight[ids[tok], :]
// B*S = 32768 tokens, D = 2048 floats (8 KB contiguous per row).
// Pure streaming gather: ~512 MiB total traffic -> HBM-bound (~22 us @ 23.3 TB/s).
//
// Design:
//  - one workgroup per token row; 256 threads (8 wave32) x 2 float4 = 2048 floats
//  - token id is wave-uniform -> scalar load + scalar base-pointer select
//  - 128-bit global loads/stores; non-temporal stores (output is write-once)

typedef float v4f __attribute__((ext_vector_type(4)));  // native clang vector: 16B

#define DIMQ4    512   // DIM / 4 float4 per row (DIM = 2048)
#define THREADS  256

__global__ __launch_bounds__(THREADS)
void embed_gather_kernel(const int* __restrict__ ids,
                         const v4f* __restrict__ weight,
                         const v4f* __restrict__ extra,
                         v4f*       __restrict__ out,
                         int over_start)
{
    const int tok = blockIdx.x;

    // Uniform across the block -> compiler emits s_load_b32 via kernarg base.
    const int id = ids[tok];
    const int oi = id - over_start;        // >= 0 iff reserved/override token

    // Scalar branchless select of the source row base (s_cselect on 64-bit base).
    const v4f* __restrict__ src =
        (oi >= 0) ? (extra  + (size_t)oi * DIMQ4)
                  : (weight + (size_t)id * DIMQ4);

    v4f* __restrict__ dst = out + (size_t)tok * DIMQ4;

    const int t = threadIdx.x;

    // Two 128-bit loads + two 128-bit NT stores per thread covers the 8 KB row.
    // Default (RT) loads: duplicated rows (reserved tokens repeat 8x) can hit L2.
    // NT stores: 256 MiB write-once output must not thrash the 192 MB L2.
    v4f a = src[t];
    v4f b = src[t + THREADS];
    __builtin_nontemporal_store(a, &dst[t]);
    __builtin_nontemporal_store(b, &dst[t + THREADS]);
}

extern "C" void kernel_launch(void* const* d_in, const int* in_sizes, int n_in,
                              void* d_out, int out_size, void* d_ws, size_t ws_size,
                              hipStream_t stream) {
    const int*   ids    = (const int*)  d_in[0];   // [B*S] int32
    const float* weight = (const float*)d_in[1];   // [VOCAB, DIM] f32
    const float* extra  = (const float*)d_in[2];   // [n_over, DIM] f32

    const int n_tok      = in_sizes[0];
    const int vocab      = in_sizes[1] / (DIMQ4 * 4);
    const int n_over     = in_sizes[2] / (DIMQ4 * 4);
    const int over_start = vocab - n_over;         // 128000 - 16 = 127984

    embed_gather_kernel<<<dim3(n_tok), dim3(THREADS), 0, stream>>>(
        ids,
        (const v4f*)weight,
        (const v4f*)extra,
        (v4f*)d_out,
        over_start);
}